// GNN_51582557224974
// MI455X (gfx1250) — compile-verified
//
#include <hip/hip_runtime.h>
#include <hip/hip_bf16.h>

// ---------------------------------------------------------------------------
// GNN: 3x GCN layers (agg = segment_sum(h[col], row); h = relu(agg@W + h@S + b))
// then out = h @ Wf^T + bf.   N=100000, D=128, E=3200000. fp32 throughout.
//
// MI455X strategy:
//  * scatter/gather dominates; h+agg (2x51.2MB) fit in 192MB L2 -> per-edge
//    wave32 gather (b128) + global_atomic_add_f32, L2-resident.
//  * GEMMs (~23 GFLOP total) via full-precision V_WMMA_F32_16X16X4_F32.
//    Register-blocked: each wave computes 5 stacked 16x16 tiles (80x16).
//    Per k-step: batch ALL fragment loads first (distinct regs), then the
//    10 WMMAs -> clause'd loads + partial s_wait_loadcnt, pipelined.
// ---------------------------------------------------------------------------

#define GNN_N 100000
#define GNN_D 128
#define GNN_E 3200000
#define MBLK  5                      // 16-row tiles per wave; 6250 = 5 * 1250

typedef __attribute__((ext_vector_type(2))) float v2f;
typedef __attribute__((ext_vector_type(8))) float v8f;

// ---------------------------------------------------------------------------
// Zero the aggregation buffer (N*D floats), vectorized b128 stores.
// ---------------------------------------------------------------------------
__global__ void gnn_zero_kernel(float4* __restrict__ p, int n4) {
    int i = blockIdx.x * blockDim.x + threadIdx.x;
    if (i < n4) p[i] = make_float4(0.f, 0.f, 0.f, 0.f);
}

// ---------------------------------------------------------------------------
// Edge scatter-add: one wave32 per edge. Lane l handles features [4l, 4l+4):
// gather 16B (global_load_b128) from h[col], 4x global_atomic_add_f32 into
// agg[row]. Both arrays are L2-resident on MI455X (192MB L2 >> 2x51.2MB).
// ---------------------------------------------------------------------------
__global__ void gnn_scatter_kernel(const float* __restrict__ h,
                                   const int* __restrict__ eidx,
                                   float* __restrict__ agg) {
    int gid  = blockIdx.x * blockDim.x + threadIdx.x;
    int e    = gid >> 5;
    int lane = gid & 31;
    if (e >= GNN_E) return;
    int row = eidx[2 * e + 0];
    int col = eidx[2 * e + 1];
    const float4 v = *(const float4*)(h + (size_t)col * GNN_D + lane * 4);
    float* dst = agg + (size_t)row * GNN_D + lane * 4;
    atomicAdd(dst + 0, v.x);
    atomicAdd(dst + 1, v.y);
    atomicAdd(dst + 2, v.z);
    atomicAdd(dst + 3, v.w);
}

// ---------------------------------------------------------------------------
// WMMA GEMM: out = (A1 @ B1) [+ A2 @ B2] + bias, optional ReLU.
// One wave computes MBLK stacked 16x16 output tiles (same 16-col strip),
// K loop in steps of 4 using V_WMMA_F32_16X16X4_F32 (full fp32).
//
// fp32 A 16x4 layout:  lane L -> row m = L&15, k-pair kb = (L>>4)*2
// fp32 B 4x16 layout:  lane L -> col n = L&15, rows kb, kb+1
// fp32 C/D 16x16:      VGPR j -> row j + 8*(L>=16), col L&15
// TRANS_B=1 means B is stored row-major [n][k] (i.e. compute A @ B^T).
// ---------------------------------------------------------------------------
template <bool TRANS_B, bool SECOND, bool RELU>
__global__ void gnn_gemm_kernel(const float* __restrict__ A1,
                                const float* __restrict__ B1,
                                const float* __restrict__ A2,
                                const float* __restrict__ B2,
                                const float* __restrict__ bias,
                                float* __restrict__ out) {
    const int lane = threadIdx.x & 31;
    const int wave = threadIdx.x >> 5;
    const int grp  = blockIdx.x * (blockDim.x >> 5) + wave;
    const int NT_N = GNN_D / 16;                       // 8 col-strips
    const int tm0  = (grp / NT_N) * (16 * MBLK);       // node-row base (80 rows)
    const int tn   = (grp % NT_N) * 16;                // feature-col base

    const int mn = lane & 15;                          // A row / B col within tile
    const int kb = (lane >> 4) << 1;                   // 0 or 2

    v8f c[MBLK];
#pragma unroll
    for (int t = 0; t < MBLK; ++t) c[t] = (v8f){};

    // Per-lane base pointers; all k-loop accesses are immediate offsets.
    const float* a1p = A1 + (size_t)(tm0 + mn) * GNN_D + kb;
    const float* a2p = SECOND ? (A2 + (size_t)(tm0 + mn) * GNN_D + kb) : nullptr;
    const float* b1c = TRANS_B ? (B1 + (size_t)(tn + mn) * GNN_D + kb)
                               : (B1 + (size_t)kb * GNN_D + tn + mn);
    const float* b2c = SECOND ? (B2 + (size_t)kb * GNN_D + tn + mn) : nullptr;

#pragma unroll
    for (int k0 = 0; k0 < GNN_D; k0 += 4) {
        // ---------- phase 1: issue ALL loads for this k-step ----------
        v2f b1f, b2f;
        v2f a1f[MBLK], a2f[MBLK];

        if (TRANS_B) {
            b1f = *(const v2f*)(b1c + k0);                       // b64
        } else {
            b1f.x = b1c[(size_t)k0 * GNN_D];                     // b32 x2, 512B apart
            b1f.y = b1c[(size_t)(k0 + 1) * GNN_D];
        }
#pragma unroll
        for (int t = 0; t < MBLK; ++t)
            a1f[t] = *(const v2f*)(a1p + (size_t)t * 16 * GNN_D + k0);   // b64

        if (SECOND) {
            b2f.x = b2c[(size_t)k0 * GNN_D];
            b2f.y = b2c[(size_t)(k0 + 1) * GNN_D];
#pragma unroll
            for (int t = 0; t < MBLK; ++t)
                a2f[t] = *(const v2f*)(a2p + (size_t)t * 16 * GNN_D + k0);
        }

        // ---------- phase 2: 10 WMMAs ----------
#pragma unroll
        for (int t = 0; t < MBLK; ++t)
            c[t] = __builtin_amdgcn_wmma_f32_16x16x4_f32(
                       false, a1f[t], false, b1f, (short)0, c[t], false, false);
        if (SECOND) {
#pragma unroll
            for (int t = 0; t < MBLK; ++t)
                c[t] = __builtin_amdgcn_wmma_f32_16x16x4_f32(
                           false, a2f[t], false, b2f, (short)0, c[t], false, false);
        }
    }

    // Epilogue: bias + optional ReLU, store per C/D layout.
    const int nc    = lane & 15;
    const int mbase = (lane >> 4) << 3;                // 0 or 8
    const float bv  = bias[tn + nc];
#pragma unroll
    for (int t = 0; t < MBLK; ++t) {
#pragma unroll
        for (int j = 0; j < 8; ++j) {
            float v = c[t][j] + bv;
            if (RELU) v = fmaxf(v, 0.f);
            out[(size_t)(tm0 + t * 16 + mbase + j) * GNN_D + tn + nc] = v;
        }
    }
}

// ---------------------------------------------------------------------------
// Host-side launch
// ---------------------------------------------------------------------------
extern "C" void kernel_launch(void* const* d_in, const int* in_sizes, int n_in,
                              void* d_out, int out_size, void* d_ws, size_t ws_size,
                              hipStream_t stream) {
    const float* x    = (const float*)d_in[0];
    const int*   eidx = (const int*)  d_in[1];
    const float* W0   = (const float*)d_in[2];
    const float* S0   = (const float*)d_in[3];
    const float* b0   = (const float*)d_in[4];
    const float* W1   = (const float*)d_in[5];
    const float* S1   = (const float*)d_in[6];
    const float* b1   = (const float*)d_in[7];
    const float* W2   = (const float*)d_in[8];
    const float* S2   = (const float*)d_in[9];
    const float* b2   = (const float*)d_in[10];
    const float* Wf   = (const float*)d_in[11];
    const float* bf   = (const float*)d_in[12];
    float* out = (float*)d_out;

    const size_t ND = (size_t)GNN_N * GNN_D;
    float* agg = (float*)d_ws;        // N*D
    float* h1  = agg + ND;            // N*D
    float* h2  = h1 + ND;             // N*D

    // Launch geometry
    const int  zeroN4 = (int)(ND / 4);                          // 3,200,000
    const dim3 zeroGrid((zeroN4 + 255) / 256);
    const dim3 scatGrid((GNN_E * 32 + 255) / 256);              // 400,000 blocks
    const int  groups = (GNN_N / (16 * MBLK)) * (GNN_D / 16);   // 1250*8 = 10,000
    const dim3 gemmGrid(groups / 4);                            // 4 waves / block
    const dim3 blk256(256), blk128(128);

    // ---- Layer 1: agg = scatter(x); h1 = relu(agg@W0 + x@S0 + b0)
    gnn_zero_kernel<<<zeroGrid, blk256, 0, stream>>>((float4*)agg, zeroN4);
    gnn_scatter_kernel<<<scatGrid, blk256, 0, stream>>>(x, eidx, agg);
    gnn_gemm_kernel<false, true, true><<<gemmGrid, blk128, 0, stream>>>(
        agg, W0, x, S0, b0, h1);

    // ---- Layer 2: agg = scatter(h1); h2 = relu(agg@W1 + h1@S1 + b1)
    gnn_zero_kernel<<<zeroGrid, blk256, 0, stream>>>((float4*)agg, zeroN4);
    gnn_scatter_kernel<<<scatGrid, blk256, 0, stream>>>(h1, eidx, agg);
    gnn_gemm_kernel<false, true, true><<<gemmGrid, blk128, 0, stream>>>(
        agg, W1, h1, S1, b1, h2);

    // ---- Layer 3: agg = scatter(h2); h1 = relu(agg@W2 + h2@S2 + b2)
    gnn_zero_kernel<<<zeroGrid, blk256, 0, stream>>>((float4*)agg, zeroN4);
    gnn_scatter_kernel<<<scatGrid, blk256, 0, stream>>>(h2, eidx, agg);
    gnn_gemm_kernel<false, true, true><<<gemmGrid, blk128, 0, stream>>>(
        agg, W2, h2, S2, b2, h1);

    // ---- Final: out = h1 @ Wf^T + bf (no relu)
    gnn_gemm_kernel<true, false, false><<<gemmGrid, blk128, 0, stream>>>(
        h1, Wf, nullptr, nullptr, bf, out);
}